// SupConLoss_64415919506184
// MI455X (gfx1250) — compile-verified
//
#include <hip/hip_runtime.h>

typedef __attribute__((ext_vector_type(2))) float v2f;
typedef __attribute__((ext_vector_type(8))) float v8f;

#define INV_T 14.285714285714286f  /* 1/0.07 */
#define NB    8192
#define ND    128
#define KSTEPS 32                   /* 128 / 4 */
#define CTILES 512                  /* 8192 / 16 */

__launch_bounds__(128, 1)
__global__ void supcon_rows_kernel(const float* __restrict__ F,
                                   const int*   __restrict__ labels,
                                   const int*   __restrict__ s_pppb,
                                   const int*   __restrict__ s_pil,
                                   const int*   __restrict__ s_fac,
                                   float*       __restrict__ row_loss)
{
    const int lane     = threadIdx.x & 31;
    const int waveId   = blockIdx.x * 4 + (threadIdx.x >> 5);
    const int row_base = waveId * 16;

    const int P   = s_pppb[0] * s_pil[0];
    const int fac = s_fac[0];

    const int half = lane >> 4;      // 0: K pair {0,1}, 1: K pair {2,3}
    const int l15  = lane & 15;      // A: row-in-tile, B: col-in-tile
    const int koff = half << 1;

    // ---- A fragments: 16 rows x 128 K, resident in registers (64 VGPR/lane)
    v2f a[KSTEPS];
    {
        const float* arow = F + (size_t)(row_base + l15) * ND + koff;
        #pragma unroll
        for (int t = 0; t < KSTEPS; ++t)
            a[t] = *(const v2f*)(arow + 4 * t);
    }

    // ---- per-row metadata for this lane's half (rows r+8*half)
    int   rlab[8];  float rscale[8];  bool rIsP[8];  bool rIsL1[8];
    #pragma unroll
    for (int r = 0; r < 8; ++r) {
        const int i = row_base + r + 8 * half;
        const int L = labels[i];
        rlab[r]   = L;
        rscale[r] = (L == fac) ? 2.0f : 1.0f;
        rIsP[r]   = (i < P);
        rIsL1[r]  = (L == 1);
    }

    // =============== PASS 1: row maxima (diagonal included, as in reference)
    float lmax[8];
    #pragma unroll
    for (int r = 0; r < 8; ++r) lmax[r] = -3.4e38f;

    for (int ct = 0; ct < CTILES; ++ct) {
        const int   j      = ct * 16 + l15;
        const int   lj     = labels[j];
        const float* brow  = F + (size_t)j * ND + koff;

        // batch all fragment loads -> one memory round-trip per tile
        v2f b[KSTEPS];
        #pragma unroll
        for (int t = 0; t < KSTEPS; ++t)
            b[t] = *(const v2f*)(brow + 4 * t);

        // prefetch next column tile while WMMAs run
        const int ctn = (ct + 1 < CTILES) ? (ct + 1) : ct;
        __builtin_prefetch(F + (size_t)(ctn * 16 + l15) * ND + koff, 0, 0);

        v8f c = {0.f,0.f,0.f,0.f,0.f,0.f,0.f,0.f};
        #pragma unroll
        for (int t = 0; t < KSTEPS; ++t)
            c = __builtin_amdgcn_wmma_f32_16x16x4_f32(
                    false, a[t], false, b[t], (short)0, c, false, false);

        const float colfac = (lj == 1) ? 0.5f : 1.0f;
        const bool  jP     = (j < P);
        #pragma unroll
        for (int r = 0; r < 8; ++r) {
            float m = INV_T;
            if (rIsP[r])        m *= colfac;
            if (rIsL1[r] && jP) m *= 0.5f;
            lmax[r] = fmaxf(lmax[r], c[r] * m);
        }
    }
    // cross-lane max within each 16-lane half
    #pragma unroll
    for (int r = 0; r < 8; ++r) {
        #pragma unroll
        for (int m = 1; m <= 8; m <<= 1)
            lmax[r] = fmaxf(lmax[r], __shfl_xor(lmax[r], m, 32));
    }

    // =============== PASS 2: per-lane partial sums (no cross-lane traffic)
    // sms/s1 are row-constant multiples of mchk/smv -> fold rscale at epilogue
    float sexp[8], smv[8], mchk[8];
    #pragma unroll
    for (int r = 0; r < 8; ++r) { sexp[r]=0.f; smv[r]=0.f; mchk[r]=0.f; }

    for (int ct = 0; ct < CTILES; ++ct) {
        const int   j      = ct * 16 + l15;
        const int   lj     = labels[j];
        const float* brow  = F + (size_t)j * ND + koff;

        v2f b[KSTEPS];
        #pragma unroll
        for (int t = 0; t < KSTEPS; ++t)
            b[t] = *(const v2f*)(brow + 4 * t);

        const int ctn = (ct + 1 < CTILES) ? (ct + 1) : ct;
        __builtin_prefetch(F + (size_t)(ctn * 16 + l15) * ND + koff, 0, 0);

        v8f c = {0.f,0.f,0.f,0.f,0.f,0.f,0.f,0.f};
        #pragma unroll
        for (int t = 0; t < KSTEPS; ++t)
            c = __builtin_amdgcn_wmma_f32_16x16x4_f32(
                    false, a[t], false, b[t], (short)0, c, false, false);

        const float colfac = (lj == 1) ? 0.5f : 1.0f;
        const bool  jP     = (j < P);
        #pragma unroll
        for (int r = 0; r < 8; ++r) {
            const int  i    = row_base + r + 8 * half;
            const bool diag = (j == i);
            float m = INV_T;
            if (rIsP[r])        m *= colfac;
            if (rIsL1[r] && jP) m *= 0.5f;
            const float val = c[r] * m;
            const float e   = __expf(val - lmax[r]);
            sexp[r] += diag ? 0.0f : e;
            const float msk = (!diag && (lj == rlab[r])) ? 1.0f : 0.0f;
            mchk[r] += msk;
            smv[r]  += msk * val;
        }
    }

    // cross-lane sum within each 16-lane half
    #pragma unroll
    for (int r = 0; r < 8; ++r) {
        #pragma unroll
        for (int m = 1; m <= 8; m <<= 1) {
            sexp[r] += __shfl_xor(sexp[r], m, 32);
            smv[r]  += __shfl_xor(smv[r],  m, 32);
            mchk[r] += __shfl_xor(mchk[r], m, 32);
        }
    }

    // loss_i = -rscale*((smv - max*mchk) - mchk*log(sexp)) / mchk
    if (l15 == 0) {
        #pragma unroll
        for (int r = 0; r < 8; ++r) {
            const float num = rscale[r] *
                ((smv[r] - lmax[r] * mchk[r]) - mchk[r] * __logf(sexp[r]));
            const float li  = (mchk[r] > 0.0f) ? (-num / mchk[r]) : 0.0f;
            row_loss[row_base + r + 8 * half] = li;
        }
    }
}

__global__ void reduce_mean_kernel(const float* __restrict__ rl,
                                   float* __restrict__ out)
{
    __shared__ float sm[256];
    float s = 0.0f;
    for (int i = threadIdx.x; i < NB; i += 256) s += rl[i];
    sm[threadIdx.x] = s;
    __syncthreads();
    for (int off = 128; off > 0; off >>= 1) {
        if (threadIdx.x < off) sm[threadIdx.x] += sm[threadIdx.x + off];
        __syncthreads();
    }
    if (threadIdx.x == 0) out[0] = sm[0] * (1.0f / (float)NB);
}

extern "C" void kernel_launch(void* const* d_in, const int* in_sizes, int n_in,
                              void* d_out, int out_size, void* d_ws, size_t ws_size,
                              hipStream_t stream) {
    const float* F      = (const float*)d_in[0];
    const int*   labels = (const int*)d_in[1];
    const int*   pppb   = (const int*)d_in[2];
    const int*   pil    = (const int*)d_in[3];
    const int*   fac    = (const int*)d_in[4];
    float*       rloss  = (float*)d_ws;

    // 512 row-blocks of 16 rows, 4 waves (128 thr) per block -> 128 blocks
    supcon_rows_kernel<<<128, 128, 0, stream>>>(F, labels, pppb, pil, fac, rloss);
    reduce_mean_kernel<<<1, 256, 0, stream>>>(rloss, (float*)d_out);
}